// SERGIOGCN_53068615910295
// MI455X (gfx1250) — compile-verified
//
#include <hip/hip_runtime.h>

typedef __attribute__((ext_vector_type(2))) float v2f;
typedef __attribute__((ext_vector_type(8))) float v8f;

#define N_NODES  200000
#define N_EDGES  6400000
#define F_OUT    128
#define N_GRAPHS 64

// Float atomic max via monotonic integer mapping (correct for +/-, init -inf).
// Lowers to native global_atomic_max_i32 / global_atomic_min_u32 (verified).
__device__ __forceinline__ void atomicMaxF32(float* addr, float v) {
    if (!(v == v)) return;                 // drop NaN
    int iv = __float_as_int(v);
    if (iv >= 0) atomicMax((int*)addr, iv);
    else         atomicMin((unsigned int*)addr, (unsigned int)iv);
}

__global__ void SERGIO_init_kernel(float* __restrict__ agg,
                                   unsigned int* __restrict__ deg,
                                   float* __restrict__ out) {
    int i = blockIdx.x * blockDim.x + threadIdx.x;
    int stride = gridDim.x * blockDim.x;
    for (int k = i; k < N_NODES; k += stride) { agg[k] = 0.0f; deg[k] = 0u; }
    for (int k = i; k < N_GRAPHS * F_OUT; k += stride) out[k] = -__builtin_inff();
}

// Phase B: edge-weighted scatter-add + degree histogram.
// Bandwidth/atomic bound: ~155 MB streaming (int64 index pairs, weights,
// gathered x) + 12.8M atomics into 1.6 MB of L2-resident accumulators.
// 2 edges per iteration -> index streams load as b128.
__global__ void SERGIO_edge_scatter_kernel(const float* __restrict__ x,
                                           const long long* __restrict__ ei,
                                           const float* __restrict__ ew,
                                           float* __restrict__ agg,
                                           unsigned int* __restrict__ deg) {
    int i = blockIdx.x * blockDim.x + threadIdx.x;
    int stride = gridDim.x * blockDim.x;
    // N_EDGES is even: exact pairs.
    for (int p = i; p < N_EDGES / 2; p += stride) {
        const int e0 = 2 * p;
        int src0 = (int)ei[e0];
        int src1 = (int)ei[e0 + 1];
        int dst0 = (int)ei[N_EDGES + e0];
        int dst1 = (int)ei[N_EDGES + e0 + 1];
        float m0 = x[src0] * ew[e0];
        float m1 = x[src1] * ew[e0 + 1];
        __hip_atomic_fetch_add(&agg[dst0], m0, __ATOMIC_RELAXED, __HIP_MEMORY_SCOPE_AGENT);
        atomicAdd(&deg[dst0], 1u);
        __hip_atomic_fetch_add(&agg[dst1], m1, __ATOMIC_RELAXED, __HIP_MEMORY_SCOPE_AGENT);
        atomicAdd(&deg[dst1], 1u);
    }
}

// Phase C: rank-3 feature expansion via V_WMMA_F32_16X16X4_F32 + segment max.
// One wave = one 16-node tile; 8 WMMAs cover F_OUT=128.
// A (16x4 f32): lanes 0-15 -> (K0,K1) = (agg_mean, x); lanes 16-31 -> (K2,K3) = (1, 0).
// B (4x16 f32): lanes 0-15 -> (K0,K1) = (W_l[f], W_r[f]); lanes 16-31 -> (K2,K3) = (b_l[f], 0).
__global__ void SERGIO_node_wmma_pool_kernel(const float* __restrict__ x,
                                             const long long* __restrict__ batch,
                                             const float* __restrict__ Wl,
                                             const float* __restrict__ bl,
                                             const float* __restrict__ Wr,
                                             const float* __restrict__ agg,
                                             const unsigned int* __restrict__ deg,
                                             float* __restrict__ out) {
    const int lane = threadIdx.x & 31;
    const int waveInBlock = threadIdx.x >> 5;
    const int tile = blockIdx.x * (blockDim.x >> 5) + waveInBlock;
    const int nTiles = N_NODES / 16;          // 12500, exact (no partial tiles)
    if (tile >= nTiles) return;               // wave-uniform exit
    const int base = tile * 16;
    const int m  = lane & 15;
    const int hi = lane >> 4;

    const float dm = (float)deg[base + m];
    const float am = agg[base + m] / fmaxf(dm, 1.0f);
    const float xm = x[base + m];
    v2f A;
    A.x = hi ? 1.0f : am;                     // K=2 row is the bias-selector "1"
    A.y = hi ? 0.0f : xm;                     // K=3 row unused

    const int gFirst = (int)batch[base];
    const int gLast  = (int)batch[base + 15];

    if (gFirst == gLast) {
        // Fast path (~99.5% of tiles): whole tile in one graph.
        const int g = gFirst;
        #pragma unroll
        for (int t = 0; t < 8; ++t) {
            const int f = t * 16 + m;
            v2f B;
            B.x = hi ? bl[f] : Wl[f];
            B.y = hi ? 0.0f  : Wr[f];
            v8f C = {};
            v8f D = __builtin_amdgcn_wmma_f32_16x16x4_f32(
                false, A, false, B, (short)0, C, false, false);
            float mx = D[0];
            #pragma unroll
            for (int j = 1; j < 8; ++j) mx = fmaxf(mx, D[j]);
            mx = fmaxf(mx, __shfl_xor(mx, 16, 32));   // merge M halves (wave32)
            if (hi == 0) atomicMaxF32(&out[g * F_OUT + f], mx);
        }
    } else {
        // Boundary tile (~64 of 12500): per-element scatter-max.
        #pragma unroll
        for (int t = 0; t < 8; ++t) {
            const int f = t * 16 + m;
            v2f B;
            B.x = hi ? bl[f] : Wl[f];
            B.y = hi ? 0.0f  : Wr[f];
            v8f C = {};
            v8f D = __builtin_amdgcn_wmma_f32_16x16x4_f32(
                false, A, false, B, (short)0, C, false, false);
            #pragma unroll
            for (int j = 0; j < 8; ++j) {
                const int node = base + j + 8 * hi;   // D VGPR j holds row M=j(+8 for hi half)
                const int g = (int)batch[node];
                atomicMaxF32(&out[g * F_OUT + f], D[j]);
            }
        }
    }
}

extern "C" void kernel_launch(void* const* d_in, const int* in_sizes, int n_in,
                              void* d_out, int out_size, void* d_ws, size_t ws_size,
                              hipStream_t stream) {
    const float*     x     = (const float*)d_in[0];
    const long long* ei    = (const long long*)d_in[1];   // int64 [2, E]
    const long long* batch = (const long long*)d_in[2];   // int64 [N], sorted
    const float*     ew    = (const float*)d_in[3];
    const float*     Wl    = (const float*)d_in[4];       // [128,1]
    const float*     bl    = (const float*)d_in[5];       // [128]
    const float*     Wr    = (const float*)d_in[6];       // [128,1]
    float* out = (float*)d_out;

    float*        agg = (float*)d_ws;                     // N_NODES floats
    unsigned int* deg = (unsigned int*)(agg + N_NODES);   // N_NODES u32

    SERGIO_init_kernel<<<512, 256, 0, stream>>>(agg, deg, out);
    SERGIO_edge_scatter_kernel<<<4096, 256, 0, stream>>>(x, ei, ew, agg, deg);

    const int nTiles = N_NODES / 16;                 // 12500 waves
    const int wavesPerBlock = 8;                     // 256 threads = 8 wave32
    const int blocks = (nTiles + wavesPerBlock - 1) / wavesPerBlock;
    SERGIO_node_wmma_pool_kernel<<<blocks, 32 * wavesPerBlock, 0, stream>>>(
        x, batch, Wl, bl, Wr, agg, deg, out);
}